// SimplePred2_33921651704224
// MI455X (gfx1250) — compile-verified
//
#include <hip/hip_runtime.h>
#include <math.h>

// Problem constants (from reference)
static constexpr int   Bn   = 1000000;
static constexpr int   NVc  = 7;
static constexpr int   EDc  = 20;
static constexpr int   NCc  = 10;
static constexpr int   IDc  = 40;      // 2*ED
static constexpr float EPSc = 1e-5f;

typedef __attribute__((ext_vector_type(2))) float v2f;
typedef __attribute__((ext_vector_type(8))) float v8f;

// ---------------------------------------------------------------------------
// Zero the per-layer batch-norm stats accumulators (4 layers x {sum[40],sq[40]})
// ---------------------------------------------------------------------------
__global__ void zero_stats_kernel(float* __restrict__ s) {
    s[threadIdx.x] = 0.0f;   // launched with exactly 4*2*ID = 320 threads
}

// ---------------------------------------------------------------------------
// Attention kernel: builds x0 = [e, att] per row, and accumulates stats of
// y0 = relu(x0 @ W0^T + b0) for the first batch-norm.
// One thread per row; 320 threads/block (1e6 / 320 = 3125 blocks, no tail).
// ---------------------------------------------------------------------------
__global__ __launch_bounds__(320) void attn_kernel(
    const int*   __restrict__ inp,
    const float* __restrict__ emb,
    const float* __restrict__ W0,
    const float* __restrict__ b0,
    float*       __restrict__ xbuf,
    float*       __restrict__ stats0)
{
    __shared__ float semb[NVc * EDc];
    __shared__ float ssum[IDc];
    __shared__ float ssq[IDc];

    const int tid = threadIdx.x;
    if (tid < NVc * EDc) semb[tid] = emb[tid];
    if (tid < IDc) { ssum[tid] = 0.0f; ssq[tid] = 0.0f; }
    __syncthreads();

    const int r    = blockIdx.x * 320 + tid;
    const int* row = inp + r * (1 + NCc);
    const int ev   = row[0];

    float e[EDc];
    #pragma unroll
    for (int d = 0; d < EDc; ++d) e[d] = semb[ev * EDc + d];

    int   ci[NCc];
    float sc[NCc];
    #pragma unroll
    for (int k = 0; k < NCc; ++k) ci[k] = row[1 + k];

    #pragma unroll
    for (int k = 0; k < NCc; ++k) {
        const float* c = semb + ci[k] * EDc;
        float s = 0.0f;
        #pragma unroll
        for (int d = 0; d < EDc; ++d) s += e[d] * c[d];
        sc[k] = s;
    }

    // softmax over the 10 scores
    float mx = sc[0];
    #pragma unroll
    for (int k = 1; k < NCc; ++k) mx = fmaxf(mx, sc[k]);
    float sum = 0.0f;
    #pragma unroll
    for (int k = 0; k < NCc; ++k) { sc[k] = expf(sc[k] - mx); sum += sc[k]; }
    const float inv = 1.0f / sum;

    float xr[IDc];
    #pragma unroll
    for (int d = 0; d < EDc; ++d) { xr[d] = e[d]; xr[EDc + d] = 0.0f; }
    #pragma unroll
    for (int k = 0; k < NCc; ++k) {
        const float wk = sc[k] * inv;
        const float* c = semb + ci[k] * EDc;
        #pragma unroll
        for (int d = 0; d < EDc; ++d) xr[EDc + d] += wk * c[d];
    }

    // Store x0 row (160 B contiguous per thread, as float4s)
    float4* dst = (float4*)(xbuf + (size_t)r * IDc);
    #pragma unroll
    for (int i = 0; i < IDc / 4; ++i)
        dst[i] = make_float4(xr[4*i], xr[4*i+1], xr[4*i+2], xr[4*i+3]);

    // Stats of y0 = relu(x0 @ W0^T + b0): wave butterfly reduce, then LDS, then global
    const int lane = tid & 31;
    for (int n = 0; n < IDc; ++n) {
        float acc = b0[n];
        #pragma unroll
        for (int d = 0; d < IDc; ++d) acc += xr[d] * W0[n * IDc + d];
        float y = fmaxf(acc, 0.0f);
        float q = y * y;
        #pragma unroll
        for (int off = 16; off > 0; off >>= 1) {
            y += __shfl_xor(y, off);
            q += __shfl_xor(q, off);
        }
        if (lane == 0) { atomicAdd(&ssum[n], y); atomicAdd(&ssq[n], q); }
    }
    __syncthreads();
    if (tid < IDc) {
        atomicAdd(&stats0[tid], ssum[tid]);
        atomicAdd(&stats0[IDc + tid], ssq[tid]);
    }
}

// ---------------------------------------------------------------------------
// WMMA layer kernel (used for layers 0..2). In-place on xbuf:
//   y_i   = relu(x @ W_i^T + b_i)           (WMMA f32 16x16x4, 3 N-tiles, K=40)
//   x     = (y_i - mean_i)*rstd_i*g_i + be_i + x
//   y_i+1 = relu(x @ W_{i+1}^T + b_{i+1})   (WMMA again, through an LDS tile)
//   accumulate column sum/sumsq of y_{i+1} into statsOut
// Block = 128 threads = 4 waves, each wave owns 16 rows. 1e6/64 = 15625 blocks.
// ---------------------------------------------------------------------------
#define WMMA_F32(ACC, A, Bv) \
    (ACC) = __builtin_amdgcn_wmma_f32_16x16x4_f32(false, (A), false, (Bv), (short)0, (ACC), false, false)

__global__ __launch_bounds__(128) void layer_kernel(
    float*       __restrict__ xbuf,
    const float* __restrict__ Wi,  const float* __restrict__ bi,
    const float* __restrict__ gi,  const float* __restrict__ bei,
    const float* __restrict__ statsIn,
    const float* __restrict__ Wn,  const float* __restrict__ bn,
    float*       __restrict__ statsOut)
{
    __shared__ float xtile[4][16][41];   // stride 41 -> no LDS bank conflicts
    __shared__ float ssum[IDc];
    __shared__ float ssq[IDc];

    const int tid = threadIdx.x;
    if (tid < IDc) { ssum[tid] = 0.0f; ssq[tid] = 0.0f; }
    __syncthreads();

    const int lane = tid & 31;
    const int w    = tid >> 5;         // wave 0..3
    const int half = lane >> 4;        // K-half selector for A/B operands
    const int ln   = lane & 15;        // = M for A-layout, = N for B/C/D layout
    const int r0   = (blockIdx.x * 4 + w) * 16;   // base row of this wave's tile

    const float invB = 1.0f / (float)Bn;

    // ---- GEMM1: y_i = x @ W_i^T  (accumulate in 3 N-tiles of 16) ----
    v8f acc[3] = {};
    for (int kk = 0; kk < 10; ++kk) {
        const int k0 = kk * 4 + 2 * half;
        const float* xr = xbuf + (size_t)(r0 + ln) * IDc + k0;
        v2f a; a.x = xr[0]; a.y = xr[1];
        #pragma unroll
        for (int t = 0; t < 3; ++t) {
            const int nG = t * 16 + ln;
            v2f b;
            if (nG < IDc) { const float* wr = Wi + nG * IDc + k0; b.x = wr[0]; b.y = wr[1]; }
            else          { b.x = 0.0f; b.y = 0.0f; }
            WMMA_F32(acc[t], a, b);
        }
    }

    // ---- bias + relu + batchnorm(statsIn) + residual; write xbuf and LDS tile ----
    #pragma unroll
    for (int t = 0; t < 3; ++t) {
        const int nG = t * 16 + ln;
        if (nG < IDc) {
            const float mean = statsIn[nG] * invB;
            const float var  = statsIn[IDc + nG] * invB - mean * mean;
            const float rstd = rsqrtf(var + EPSc);
            const float g    = gi[nG];
            const float be   = bei[nG];
            const float bias = bi[nG];
            #pragma unroll
            for (int v = 0; v < 8; ++v) {
                const int m   = v + 8 * half;        // C/D layout row
                const int row = r0 + m;
                float y  = fmaxf(acc[t][v] + bias, 0.0f);
                y        = (y - mean) * rstd * g + be;
                float xn = y + xbuf[(size_t)row * IDc + nG];
                xbuf[(size_t)row * IDc + nG] = xn;
                xtile[w][m][nG] = xn;
            }
        }
    }
    __syncthreads();   // makes the C/D->A relayout through LDS airtight

    // ---- GEMM2: y_{i+1} = relu(x_new @ W_{i+1}^T + b_{i+1}) for stats ----
    v8f acc2[3] = {};
    for (int kk = 0; kk < 10; ++kk) {
        const int k0 = kk * 4 + 2 * half;
        v2f a; a.x = xtile[w][ln][k0]; a.y = xtile[w][ln][k0 + 1];
        #pragma unroll
        for (int t = 0; t < 3; ++t) {
            const int nG = t * 16 + ln;
            v2f b;
            if (nG < IDc) { const float* wr = Wn + nG * IDc + k0; b.x = wr[0]; b.y = wr[1]; }
            else          { b.x = 0.0f; b.y = 0.0f; }
            WMMA_F32(acc2[t], a, b);
        }
    }

    #pragma unroll
    for (int t = 0; t < 3; ++t) {
        const int nG = t * 16 + ln;
        if (nG < IDc) {
            const float bias = bn[nG];
            float s = 0.0f, q = 0.0f;
            #pragma unroll
            for (int v = 0; v < 8; ++v) {
                const float y = fmaxf(acc2[t][v] + bias, 0.0f);
                s += y; q += y * y;
            }
            atomicAdd(&ssum[nG], s);
            atomicAdd(&ssq[nG], q);
        }
    }
    __syncthreads();
    if (tid < IDc) {
        atomicAdd(&statsOut[tid],       ssum[tid]);
        atomicAdd(&statsOut[IDc + tid], ssq[tid]);
    }
}

// ---------------------------------------------------------------------------
// Final kernel: recompute y3 from x3 (stats3 ready), normalize, residual,
// dot with Wf, sigmoid. One thread per row.
// ---------------------------------------------------------------------------
__global__ __launch_bounds__(320) void final_kernel(
    const float* __restrict__ xbuf,
    const float* __restrict__ W3, const float* __restrict__ b3,
    const float* __restrict__ g3, const float* __restrict__ be3,
    const float* __restrict__ stats3,
    const float* __restrict__ Wf, const float* __restrict__ bf,
    float*       __restrict__ out)
{
    const int r = blockIdx.x * 320 + threadIdx.x;
    float x[IDc];
    const float4* src = (const float4*)(xbuf + (size_t)r * IDc);
    #pragma unroll
    for (int i = 0; i < IDc / 4; ++i) {
        float4 v = src[i];
        x[4*i] = v.x; x[4*i+1] = v.y; x[4*i+2] = v.z; x[4*i+3] = v.w;
    }
    const float invB = 1.0f / (float)Bn;
    float o = bf[0];
    for (int n = 0; n < IDc; ++n) {
        float acc = b3[n];
        #pragma unroll
        for (int d = 0; d < IDc; ++d) acc += x[d] * W3[n * IDc + d];
        const float y    = fmaxf(acc, 0.0f);
        const float mean = stats3[n] * invB;
        const float var  = stats3[IDc + n] * invB - mean * mean;
        const float x4   = (y - mean) * rsqrtf(var + EPSc) * g3[n] + be3[n] + x[n];
        o += x4 * Wf[n];
    }
    out[r] = 1.0f / (1.0f + expf(-o));
}

// ---------------------------------------------------------------------------
extern "C" void kernel_launch(void* const* d_in, const int* in_sizes, int n_in,
                              void* d_out, int out_size, void* d_ws, size_t ws_size,
                              hipStream_t stream) {
    const int*   inp    = (const int*)  d_in[0];
    const float* emb    = (const float*)d_in[1];
    const float* Ws     = (const float*)d_in[2];
    const float* bs     = (const float*)d_in[3];
    const float* gammas = (const float*)d_in[4];
    const float* betas  = (const float*)d_in[5];
    const float* Wf     = (const float*)d_in[6];
    const float* bf     = (const float*)d_in[7];
    float* out  = (float*)d_out;

    // Workspace layout: [ xbuf: B*40 f32 (160 MB) ][ stats: 4 layers * 80 f32 ]
    float* xbuf  = (float*)d_ws;
    float* stats = xbuf + (size_t)Bn * IDc;

    zero_stats_kernel<<<1, 4 * 2 * IDc, 0, stream>>>(stats);

    attn_kernel<<<Bn / 320, 320, 0, stream>>>(inp, emb, Ws, bs, xbuf, stats);

    for (int i = 0; i < 3; ++i) {
        layer_kernel<<<Bn / 64, 128, 0, stream>>>(
            xbuf,
            Ws + i * IDc * IDc,       bs + i * IDc,
            gammas + i * IDc,         betas + i * IDc,
            stats + i * 2 * IDc,
            Ws + (i + 1) * IDc * IDc, bs + (i + 1) * IDc,
            stats + (i + 1) * 2 * IDc);
    }

    final_kernel<<<Bn / 320, 320, 0, stream>>>(
        xbuf,
        Ws + 3 * IDc * IDc, bs + 3 * IDc,
        gammas + 3 * IDc,   betas + 3 * IDc,
        stats + 3 * 2 * IDc,
        Wf, bf, out);
}